// GraphSAGE_15779709845831
// MI455X (gfx1250) — compile-verified
//
#include <hip/hip_runtime.h>
#include <hip/hip_bf16.h>

typedef __attribute__((ext_vector_type(2))) float v2f;
typedef __attribute__((ext_vector_type(8))) float v8f;

#define WSTRIDE 66   // 64 + 2 pad: avoids 16-way LDS bank conflicts on column reads

// ---------------------------------------------------------------------------
// Zero a float buffer
// ---------------------------------------------------------------------------
__global__ void zero_f32(float* __restrict__ p, long long n) {
    long long i = (long long)blockIdx.x * blockDim.x + threadIdx.x;
    if (i < n) p[i] = 0.0f;
}

// ---------------------------------------------------------------------------
// Per-edge degree accumulation: deg[col[e]] += 1
// ---------------------------------------------------------------------------
__global__ void deg_kernel(const int* __restrict__ col, float* __restrict__ deg, int E) {
    int e = blockIdx.x * blockDim.x + threadIdx.x;
    if (e < E) atomicAdd(&deg[col[e]], 1.0f);
}

// ---------------------------------------------------------------------------
// Scatter-add: agg[col[e]][:] += src[row[e]][:]
// 16 threads per edge, float4 chunk each. src/agg are L2-resident (25.6 MB).
// ---------------------------------------------------------------------------
__global__ void scatter_add(const float* __restrict__ src, const int* __restrict__ row,
                            const int* __restrict__ col, float* __restrict__ agg, int E) {
    long long gid = (long long)blockIdx.x * blockDim.x + threadIdx.x;
    int e = (int)(gid >> 4);
    int c = (int)(gid & 15);
    if (e >= E) return;
    int r = row[e];
    int d = col[e];
    const float4 v = *(const float4*)(src + (long long)r * 64 + c * 4);
    float* dst = agg + (long long)d * 64 + c * 4;
    atomicAdd(dst + 0, v.x);
    atomicAdd(dst + 1, v.y);
    atomicAdd(dst + 2, v.z);
    atomicAdd(dst + 3, v.w);
}

// ---------------------------------------------------------------------------
// Fused SAGEConv linear:  out = (agg/max(deg,1)) @ Wl^T + bl + xin @ Wr^T
// One wave = one 16-node tile. 4 output col-tiles, K=64 via 16x v_wmma_f32_16x16x4_f32
// per (col-tile, weight matrix). Weights staged in LDS with padded stride.
//
// f32 WMMA VGPR layouts (ISA 7.12.2):
//   A 16x4:  lane m = lane&15, half = lane>>4, a[i] = A[m][2*half + i]
//   B 4x16:  lane n = lane&15,                b[i] = B[2*half + i][n]
//   C/D 16x16: lane n = lane&15, acc[v] = D[8*half + v][n]
// ---------------------------------------------------------------------------
__global__ __launch_bounds__(256) void sage_linear_wmma(
    const float* __restrict__ agg, const float* __restrict__ deg,
    const float* __restrict__ xin,
    const float* __restrict__ Wl, const float* __restrict__ bl,
    const float* __restrict__ Wr,
    float* __restrict__ out, int n_nodes)
{
    __shared__ float sWl[64 * WSTRIDE];
    __shared__ float sWr[64 * WSTRIDE];
    __shared__ float sbl[64];

    const int tid = threadIdx.x;

    // Stage weights: W[o][i] (row-major 64x64) -> sW[o*WSTRIDE + i]
    for (int idx = tid; idx < 64 * 64; idx += 256) {
        int o = idx >> 6, i = idx & 63;
        sWl[o * WSTRIDE + i] = Wl[idx];
        sWr[o * WSTRIDE + i] = Wr[idx];
    }
    if (tid < 64) sbl[tid] = bl[tid];
    __syncthreads();

    const int wave = tid >> 5;
    const int lane = tid & 31;
    const int tile = blockIdx.x * 8 + wave;          // 16-node tile per wave
    if (tile * 16 >= n_nodes) return;                // whole-wave uniform exit

    const int m    = lane & 15;
    const int half = lane >> 4;
    const int node = tile * 16 + m;

    const float rdeg = 1.0f / fmaxf(deg[node], 1.0f);

    const float* aggRow = agg + node * 64;
    const float* xRow   = xin + node * 64;

    v8f acc0 = {}, acc1 = {}, acc2 = {}, acc3 = {};
    v8f accs[4];

    for (int t = 0; t < 16; ++t) {
        const int kb = t * 4 + half * 2;             // K base for this fragment

        v2f a_m;                                     // mean fragment (agg * 1/deg)
        a_m.x = aggRow[kb]     * rdeg;
        a_m.y = aggRow[kb + 1] * rdeg;
        const v2f a_x = *(const v2f*)(xRow + kb);    // self fragment

        #pragma unroll
        for (int nt = 0; nt < 4; ++nt) {
            const int nc = nt * 16 + m;              // output column
            const v2f bL = *(const v2f*)(&sWl[nc * WSTRIDE + kb]);
            const v2f bR = *(const v2f*)(&sWr[nc * WSTRIDE + kb]);
            v8f& acc = (nt == 0) ? acc0 : (nt == 1) ? acc1 : (nt == 2) ? acc2 : acc3;
            acc = __builtin_amdgcn_wmma_f32_16x16x4_f32(
                      false, a_m, false, bL, (short)0, acc, false, false);
            acc = __builtin_amdgcn_wmma_f32_16x16x4_f32(
                      false, a_x, false, bR, (short)0, acc, false, false);
        }
    }

    accs[0] = acc0; accs[1] = acc1; accs[2] = acc2; accs[3] = acc3;

    // Bias + store. Lanes 0-15 cover rows tile*16+v, lanes 16-31 rows +8.
    #pragma unroll
    for (int nt = 0; nt < 4; ++nt) {
        const int col = nt * 16 + m;
        const float b = sbl[col];
        #pragma unroll
        for (int v = 0; v < 8; ++v) {
            const int r = tile * 16 + half * 8 + v;
            out[(long long)r * 64 + col] = accs[nt][v] + b;
        }
    }
}

// ---------------------------------------------------------------------------
// Edge scorer: out[e] = fc_b + sum_f h[row][f] * h[col][f] * fc_w[f]
// ---------------------------------------------------------------------------
__global__ void edge_score(const float* __restrict__ h, const int* __restrict__ row,
                           const int* __restrict__ col, const float* __restrict__ fcw,
                           const float* __restrict__ fcb, float* __restrict__ out, int E) {
    int e = blockIdx.x * blockDim.x + threadIdx.x;
    if (e >= E) return;
    const float4* hr = (const float4*)(h + (long long)row[e] * 64);
    const float4* hc = (const float4*)(h + (long long)col[e] * 64);
    const float4* w4 = (const float4*)fcw;
    float s = 0.0f;
    #pragma unroll
    for (int i = 0; i < 16; ++i) {
        const float4 a = hr[i];
        const float4 b = hc[i];
        const float4 w = w4[i];
        s += a.x * b.x * w.x + a.y * b.y * w.y + a.z * b.z * w.z + a.w * b.w * w.w;
    }
    out[e] = s + fcb[0];
}

// ---------------------------------------------------------------------------
// Launch
// ---------------------------------------------------------------------------
extern "C" void kernel_launch(void* const* d_in, const int* in_sizes, int n_in,
                              void* d_out, int out_size, void* d_ws, size_t ws_size,
                              hipStream_t stream) {
    const float* x   = (const float*)d_in[0];
    const int*   ei  = (const int*)d_in[1];   // [2][E] int32
    const float* Wl0 = (const float*)d_in[2];
    const float* bl0 = (const float*)d_in[3];
    const float* Wr0 = (const float*)d_in[4];
    const float* Wl1 = (const float*)d_in[5];
    const float* bl1 = (const float*)d_in[6];
    const float* Wr1 = (const float*)d_in[7];
    const float* fcw = (const float*)d_in[8];
    const float* fcb = (const float*)d_in[9];
    float* out = (float*)d_out;

    const long long N = in_sizes[0] / 64;
    const long long E = in_sizes[1] / 2;
    const int* row = ei;
    const int* col = ei + E;

    float* ws  = (float*)d_ws;
    float* agg = ws;                 // N*64
    float* deg = agg + N * 64;       // N
    float* h0  = deg + N;            // N*64
    float* h1  = h0 + N * 64;        // N*64

    const int TB = 256;
    const long long zn     = N * 64 + N;              // agg + deg contiguous
    const long long sgrid  = (E * 16 + TB - 1) / TB;  // scatter threads
    const int       ntiles = (int)((N + 15) / 16);
    const int       lgrid  = (ntiles + 7) / 8;

    // ---- Layer 0 ----
    zero_f32<<<dim3((unsigned)((zn + TB - 1) / TB)), dim3(TB), 0, stream>>>(agg, zn);
    deg_kernel<<<dim3((unsigned)((E + TB - 1) / TB)), dim3(TB), 0, stream>>>(col, deg, (int)E);
    scatter_add<<<dim3((unsigned)sgrid), dim3(TB), 0, stream>>>(x, row, col, agg, (int)E);
    sage_linear_wmma<<<dim3(lgrid), dim3(256), 0, stream>>>(agg, deg, x, Wl0, bl0, Wr0, h0, (int)N);

    // ---- Layer 1 (deg unchanged; re-zero agg only) ----
    zero_f32<<<dim3((unsigned)((N * 64 + TB - 1) / TB)), dim3(TB), 0, stream>>>(agg, N * 64);
    scatter_add<<<dim3((unsigned)sgrid), dim3(TB), 0, stream>>>(h0, row, col, agg, (int)E);
    sage_linear_wmma<<<dim3(lgrid), dim3(256), 0, stream>>>(agg, deg, h0, Wl1, bl1, Wr1, h1, (int)N);

    // ---- Edge scoring ----
    edge_score<<<dim3((unsigned)((E + TB - 1) / TB)), dim3(TB), 0, stream>>>(h1, row, col, fcw, fcb, out, (int)E);
}